// SeriesMemoryBank_71622874628138
// MI455X (gfx1250) — compile-verified
//
#include <hip/hip_runtime.h>
#include <math.h>

// Problem constants (match reference setup_inputs)
#define BQ     2048
#define NM     50000
#define DIM    512
#define TOPK   16
#define NSPLIT 8                 // grid.y splits over memory tiles
#define LISTS  (NSPLIT * 4)      // 4 waves per block -> 32 partial lists per query
#define NTILES (NM / 16)         // 3125

typedef __attribute__((ext_vector_type(2))) float v2f;
typedef __attribute__((ext_vector_type(8))) float v8f;

// ---------------------------------------------------------------------------
// Kernel 0: inverse L2 norms for memory rows and query rows; zero the
// "any_valid" flag. One wave32 per row, butterfly shuffle reduction.
// ---------------------------------------------------------------------------
__global__ void smb_norms(const float* __restrict__ mem,
                          const float* __restrict__ qry,
                          float* __restrict__ invm,
                          float* __restrict__ invq,
                          int* __restrict__ flag) {
  if (blockIdx.x == 0 && threadIdx.x == 0) *flag = 0;
  int wid  = (int)((blockIdx.x * blockDim.x + threadIdx.x) >> 5);
  int lane = threadIdx.x & 31;
  if (wid >= NM + BQ) return;
  const float* row = (wid < NM) ? (mem + (size_t)wid * DIM)
                                : (qry + (size_t)(wid - NM) * DIM);
  float ss = 0.f;
  for (int c = lane; c < DIM; c += 32) { float x = row[c]; ss += x * x; }
#pragma unroll
  for (int m = 16; m >= 1; m >>= 1) ss += __shfl_xor(ss, m, 32);
  if (lane == 0) {
    float inv = 1.0f / fmaxf(sqrtf(ss), 1e-12f);
    if (wid < NM) invm[wid] = inv; else invq[wid - NM] = inv;
  }
}

// ---------------------------------------------------------------------------
// Kernel 1: fp32 WMMA similarity + streaming per-wave top-K.
// Block = 4 waves sharing a 16-query A tile in LDS. Each wave owns candidate
// list L = blockIdx.y*4 + wave and walks memory tiles t = L, L+32, ...
// Simple per-step operand loads: the gfx1250 scheduler software-pipelines
// the staging global loads under the WMMA stream and multi-buffers the
// ds_load operands on its own (verified in round-1 disassembly). No register
// cap.
// ---------------------------------------------------------------------------
__global__ void
__launch_bounds__(128)
smb_sims_topk(const float* __restrict__ qry,
              const int*   __restrict__ qids,
              const float* __restrict__ mem,
              const int*   __restrict__ mids,
              const float* __restrict__ invm,
              const float* __restrict__ invq,
              float* __restrict__ candV,
              int*   __restrict__ candI,
              int*   __restrict__ flag) {
  __shared__ float As[16 * 516];        // query tile, padded stride (bank-safe)
  __shared__ float Bs[4][16 * 68];      // per-wave memory K-slice, padded
  __shared__ float Ss[4][16 * 17];      // per-wave 16x16 sims, padded
  __shared__ float invqS[16];
  __shared__ int   qidS[16];
  __shared__ float invmS[4][16];
  __shared__ int   midS[4][16];

  const int qbase = blockIdx.x * 16;
  const int tid   = threadIdx.x;
  const int wave  = tid >> 5;
  const int lane  = tid & 31;
  const int hi    = lane >> 4;   // half-wave select (+2 K offset per ISA layout)
  const int lo    = lane & 15;

  // Stage 16 query rows (coalesced), plus their inv-norms and ids.
  for (int idx = tid; idx < 16 * DIM; idx += 128) {
    int r = idx >> 9, c = idx & (DIM - 1);
    As[r * 516 + c] = qry[(size_t)(qbase + r) * DIM + c];
  }
  if (tid < 16) { invqS[tid] = invq[qbase + tid]; qidS[tid] = qids[qbase + tid]; }
  __syncthreads();

  const int L    = blockIdx.y * 4 + wave;
  const int aoff = lo * 516 + 2 * hi;   // A operand base col offset
  const int boff = lo * 68 + 2 * hi;    // B operand base col offset (in slice)

  float topv[TOPK];
  int   topi[TOPK];
#pragma unroll
  for (int j = 0; j < TOPK; ++j) { topv[j] = -INFINITY; topi[j] = -1; }

  for (int t = L; t < NTILES; t += LISTS) {
    const int nbase = t * 16;
    if (lane < 16) {
      invmS[wave][lane] = invm[nbase + lane];
      midS[wave][lane]  = mids[nbase + lane];
    }
    const float* tilePtr = mem + (size_t)nbase * DIM;
    v8f acc = {0.f, 0.f, 0.f, 0.f, 0.f, 0.f, 0.f, 0.f};

    for (int ks = 0; ks < DIM; ks += 64) {
      // Stage B slice: 16 rows x 64 cols = 256 float4 quads (8 per lane),
      // coalesced global b128 loads into padded LDS.
#pragma unroll
      for (int j = 0; j < 8; ++j) {
        int qd = lane + 32 * j;                    // 0..255 quads
        int r  = qd >> 4;                          // 0..15 rows
        int c4 = (qd & 15) * 4;                    // 0..60 col
        const float4 v = *(const float4*)&tilePtr[(size_t)r * DIM + ks + c4];
        *(float4*)&Bs[wave][r * 68 + c4] = v;
      }
      // 16 WMMA K-steps over this slice (K=4 each).
#pragma unroll
      for (int kk = 0; kk < 16; ++kk) {
        v2f a = *(const v2f*)&As[aoff + ks + kk * 4];
        v2f b = *(const v2f*)&Bs[wave][boff + kk * 4];
        acc = __builtin_amdgcn_wmma_f32_16x16x4_f32(
            false, a, false, b, (short)0, acc, false, false);
      }
    }

    // Scale raw dot products by both inverse norms; dump 16x16 sims to LDS.
    {
      float sm = invmS[wave][lo];
#pragma unroll
      for (int v = 0; v < 8; ++v) {
        int row = v + 8 * hi;                       // C layout: M = v + 8*hi
        Ss[wave][row * 17 + lo] = acc[v] * invqS[row] * sm;
      }
    }

    // Streaming top-K: lane q (0..15) owns query row q.
    if (lane < 16) {
      const int myqid = qidS[lane];
#pragma unroll 1
      for (int j = 0; j < 16; ++j) {
        float s  = Ss[wave][lane * 17 + j];
        bool  ok = (s >= 0.0f) && (midS[wave][j] != myqid);
        if (ok && s > topv[TOPK - 1]) {
          float cv = s; int ci = nbase + j;
#pragma unroll
          for (int p = 0; p < TOPK; ++p) {
            bool  sw = cv > topv[p];
            float tv = topv[p]; int ti = topi[p];
            topv[p] = sw ? cv : tv;  topi[p] = sw ? ci : ti;
            cv      = sw ? tv : cv;  ci      = sw ? ti : ci;
          }
        }
      }
    }
  }

  // Emit this wave's sorted candidate list.
  if (lane < 16) {
    const size_t base = ((size_t)(qbase + lane) * LISTS + L) * TOPK;
#pragma unroll
    for (int j = 0; j < TOPK; ++j) { candV[base + j] = topv[j]; candI[base + j] = topi[j]; }
    if (topv[0] >= 0.0f) *flag = 1;   // benign same-value race
  }
}

// ---------------------------------------------------------------------------
// Kernel 2: per-query merge of 32 sorted lists (512 candidates) via exact
// rank-by-counting (stable ties by index), then cooperative row gather.
// ---------------------------------------------------------------------------
__launch_bounds__(256)
__global__ void smb_merge(const float* __restrict__ candV,
                          const int*   __restrict__ candI,
                          const float* __restrict__ mem,
                          const int*   __restrict__ mids,
                          const int*   __restrict__ flag,
                          float* __restrict__ out) {
  const int NC = LISTS * TOPK;          // 512
  __shared__ float cv[LISTS * TOPK];
  __shared__ int   ci[LISTS * TOPK];
  __shared__ float sv[TOPK];
  __shared__ int   si[TOPK];

  const int q = blockIdx.x;
  const int t = threadIdx.x;

  if (t < TOPK) { sv[t] = -INFINITY; si[t] = -1; }
  for (int i = t; i < NC; i += 256) {
    cv[i] = candV[(size_t)q * NC + i];
    ci[i] = candI[(size_t)q * NC + i];
  }
  __syncthreads();

  for (int i = t; i < NC; i += 256) {
    float v = cv[i]; int id = ci[i];
    if (v > -INFINITY) {
      int rank = 0;
      for (int j = 0; j < NC; ++j) {
        float vj = cv[j];
        rank += (vj > v) || (vj == v && ci[j] < id);
      }
      if (rank < TOPK) { sv[rank] = v; si[rank] = id; }
    }
  }
  __syncthreads();

  const int any = *flag;
  float* outR = out;                                  // retrieved (B,K,D)
  float* outS = out + (size_t)BQ * TOPK * DIM;        // top_sims  (B,K)
  float* outM = outS + (size_t)BQ * TOPK;             // mask      (B,K)
  float* outI = outM + (size_t)BQ * TOPK;             // ids       (B,K)

  if (t < TOPK) {
    float s = sv[t]; int id = si[t];
    bool  m = (s > -INFINITY) && (any != 0);
    outS[(size_t)q * TOPK + t] = any ? s : 0.0f;
    outM[(size_t)q * TOPK + t] = m ? 1.0f : 0.0f;
    int rid = (any && id >= 0) ? mids[id] : -1;
    outI[(size_t)q * TOPK + t] = (float)rid;
  }
  for (int f = t; f < TOPK * DIM; f += 256) {
    int j  = f >> 9;                  // DIM = 512
    int c  = f & (DIM - 1);
    int id = si[j];
    float val = (any && id >= 0) ? mem[(size_t)id * DIM + c] : 0.0f;
    outR[(size_t)q * TOPK * DIM + f] = val;
  }
}

// ---------------------------------------------------------------------------
extern "C" void kernel_launch(void* const* d_in, const int* in_sizes, int n_in,
                              void* d_out, int out_size, void* d_ws, size_t ws_size,
                              hipStream_t stream) {
  const float* qrep = (const float*)d_in[0];   // (B,D) fp32
  const int*   qids = (const int*)d_in[1];     // (B,)  int32
  const float* mem  = (const float*)d_in[2];   // (N,D) fp32
  const int*   mids = (const int*)d_in[3];     // (N,)  int32
  float* out = (float*)d_out;
  float* ws  = (float*)d_ws;

  // Workspace layout (floats): invm[N] | invq[B] | flag | pad | candV | candI
  float* invm  = ws;
  float* invq  = ws + NM;
  int*   flag  = (int*)(ws + NM + BQ);
  float* candV = ws + (((NM + BQ) + 63) & ~63);              // 64-elt aligned
  int*   candI = (int*)(candV + (size_t)BQ * LISTS * TOPK);

  // Kernel 0: inverse norms (one wave per row; 8 waves per 256-thread block)
  int nrows = NM + BQ;
  smb_norms<<<(nrows + 7) / 8, 256, 0, stream>>>(mem, qrep, invm, invq, flag);

  // Kernel 1: WMMA sims + streaming per-wave top-K
  smb_sims_topk<<<dim3(BQ / 16, NSPLIT), 128, 0, stream>>>(
      qrep, qids, mem, mids, invm, invq, candV, candI, flag);

  // Kernel 2: merge + gather + write all outputs
  smb_merge<<<BQ, 256, 0, stream>>>(candV, candI, mem, mids, flag, out);
}